// United_Model_91259465105608
// MI455X (gfx1250) — compile-verified
//
#include <hip/hip_runtime.h>

#define B_   8
#define T_   64
#define N_   200
#define CIN_ 200
#define C_   128
#define L_   2
#define NC_  2
#define S_   100
#define NN_  (N_*N_)          // 40000
#define RTOK_ (B_*T_*N_)      // 102400  (multiple of 32)
#define RTB_  (T_*B_)         // 512     (multiple of 32)

// output layout (flat, concatenated in reference return order)
#define OUT_LOGIT 0                       // 16
#define OUT_FFMRI 16                      // 1024
#define OUT_FSMRI 1040                    // 1024
#define OUT_NODE  2064                    // 8*2*64*200 = 204800
#define OUT_TIME  206864                  // 8*2*64*64  = 65536

typedef float v2f  __attribute__((ext_vector_type(2)));
typedef float v8f  __attribute__((ext_vector_type(8)));

__device__ __forceinline__ v8f wmma_f32(v2f a, v2f b, v8f c) {
  return __builtin_amdgcn_wmma_f32_16x16x4_f32(false, a, false, b, (short)0, c,
                                               false, false);
}

// ---------------------------------------------------------------------------
// WMMA GEMM, compile-time K/Ncol/ACT: D[r,n] = act( sum_k A[r,k]*W[k,n] + b[n] )
// One wave computes a 32x64 tile: 2 A fragments x 4 B fragments -> 8 WMMAs
// per K-step. Requirements: R % 32 == 0, K % 4 == 0 (true at every call site).
// Column edge (Ncol=200) handled by clamped loads + masked stores.
// ACT: 0=none, 1=relu, 2=sigmoid
// ---------------------------------------------------------------------------
template<int K, int Ncol, int ACT>
__global__ __launch_bounds__(32)
void gemm_wmma(const float* __restrict__ A, const float* __restrict__ W,
               const float* __restrict__ bias, float* __restrict__ D, int R)
{
  const int lane = threadIdx.x & 31;
  const int half = lane >> 4;        // 0 or 1
  const int l16  = lane & 15;
  const int col0 = blockIdx.x * 64;
  const int row0 = blockIdx.y * 32;

  const float* Ar0 = A + (size_t)(row0 + l16) * K + (half << 1);
  const float* Ar1 = Ar0 + (size_t)16 * K;

  int  nc[4];
  bool nv[4];
#pragma unroll
  for (int j = 0; j < 4; ++j) {
    int n = col0 + j * 16 + l16;
    nv[j] = (n < Ncol);
    nc[j] = nv[j] ? n : (Ncol - 1);   // clamp: garbage only hits masked column
  }
  const float* Wl0 = W + (size_t)(half << 1) * Ncol + nc[0];
  const float* Wl1 = W + (size_t)(half << 1) * Ncol + nc[1];
  const float* Wl2 = W + (size_t)(half << 1) * Ncol + nc[2];
  const float* Wl3 = W + (size_t)(half << 1) * Ncol + nc[3];

  v8f acc[2][4];
#pragma unroll
  for (int jr = 0; jr < 2; ++jr)
#pragma unroll
    for (int j = 0; j < 4; ++j)
      acc[jr][j] = (v8f){0,0,0,0,0,0,0,0};

#pragma unroll 2
  for (int k0 = 0; k0 < K; k0 += 4) {
    v2f a0 = *(const v2f*)Ar0;                 // aligned b64 loads
    v2f a1 = *(const v2f*)Ar1;
    Ar0 += 4; Ar1 += 4;
    v2f b0, b1, b2, b3;
    b0.x = Wl0[0]; b0.y = Wl0[Ncol];           // immediate-offset b32 loads
    b1.x = Wl1[0]; b1.y = Wl1[Ncol];
    b2.x = Wl2[0]; b2.y = Wl2[Ncol];
    b3.x = Wl3[0]; b3.y = Wl3[Ncol];
    Wl0 += 4 * Ncol; Wl1 += 4 * Ncol; Wl2 += 4 * Ncol; Wl3 += 4 * Ncol;
    acc[0][0] = wmma_f32(a0, b0, acc[0][0]);
    acc[0][1] = wmma_f32(a0, b1, acc[0][1]);
    acc[0][2] = wmma_f32(a0, b2, acc[0][2]);
    acc[0][3] = wmma_f32(a0, b3, acc[0][3]);
    acc[1][0] = wmma_f32(a1, b0, acc[1][0]);
    acc[1][1] = wmma_f32(a1, b1, acc[1][1]);
    acc[1][2] = wmma_f32(a1, b2, acc[1][2]);
    acc[1][3] = wmma_f32(a1, b3, acc[1][3]);
  }

#pragma unroll
  for (int jr = 0; jr < 2; ++jr) {
#pragma unroll
    for (int j = 0; j < 4; ++j) {
      if (!nv[j]) continue;
      const int n = col0 + j * 16 + l16;
      const float bv = bias ? bias[n] : 0.f;
#pragma unroll
      for (int i = 0; i < 8; ++i) {
        int mr = row0 + jr * 16 + i + (half << 3);
        float v = acc[jr][j][i] + bv;
        if (ACT == 1)      v = fmaxf(v, 0.f);
        else if (ACT == 2) v = 1.f / (1.f + __expf(-v));
        D[(size_t)mr * Ncol + n] = v;
      }
    }
  }
}

// ---------------------------------------------------------------------------
// agg = (mask + eps*I) @ h3 per (b,t) batch (batched 200x200x128), where
// mask = (adj > thr). The eps*h3 term is folded in as a diagonal on A.
// One wave computes a 32x64 tile; rows clamped (N_=200), cols always valid.
// ---------------------------------------------------------------------------
__global__ __launch_bounds__(32)
void agg_wmma(const float* __restrict__ adj, const float* __restrict__ thr,
              const float* __restrict__ h3, const float* __restrict__ eps_arr,
              int layer, float* __restrict__ out)
{
  const int bt   = blockIdx.z;
  const int lane = threadIdx.x & 31;
  const int half = lane >> 4;
  const int l16  = lane & 15;
  const int row0 = blockIdx.y * 32;   // node i (up to 223, clamp)
  const int col0 = blockIdx.x * 64;   // channel (always < 128)
  const float th  = thr[bt];
  const float eps = eps_arr[layer];
  const float* a_bt = adj + (size_t)bt * NN_;
  const float* h_bt = h3  + (size_t)bt * N_ * C_;

  const int m0 = row0 + l16;
  const int m1 = m0 + 16;
  const int mc0 = (m0 < N_) ? m0 : (N_ - 1);
  const int mc1 = (m1 < N_) ? m1 : (N_ - 1);
  const int kb = (half << 1);
  const float* Ar0 = a_bt + (size_t)mc0 * N_ + kb;
  const float* Ar1 = a_bt + (size_t)mc1 * N_ + kb;
  const float* Hl  = h_bt + (size_t)kb * C_ + col0 + l16;

  v8f acc[2][4];
#pragma unroll
  for (int jr = 0; jr < 2; ++jr)
#pragma unroll
    for (int j = 0; j < 4; ++j)
      acc[jr][j] = (v8f){0,0,0,0,0,0,0,0};

#pragma unroll 2
  for (int k0 = 0; k0 < N_; k0 += 4) {
    const int k = k0 + kb;
    v2f r0 = *(const v2f*)Ar0;                 // aligned b64 loads
    v2f r1 = *(const v2f*)Ar1;
    Ar0 += 4; Ar1 += 4;
    v2f a0, a1;
    a0.x = ((r0.x > th) ? 1.f : 0.f) + ((k     == m0) ? eps : 0.f);
    a0.y = ((r0.y > th) ? 1.f : 0.f) + ((k + 1 == m0) ? eps : 0.f);
    a1.x = ((r1.x > th) ? 1.f : 0.f) + ((k     == m1) ? eps : 0.f);
    a1.y = ((r1.y > th) ? 1.f : 0.f) + ((k + 1 == m1) ? eps : 0.f);
    v2f b0, b1, b2, b3;
    b0.x = Hl[0];  b0.y = Hl[C_ + 0];          // immediate-offset b32 loads
    b1.x = Hl[16]; b1.y = Hl[C_ + 16];
    b2.x = Hl[32]; b2.y = Hl[C_ + 32];
    b3.x = Hl[48]; b3.y = Hl[C_ + 48];
    Hl += 4 * C_;
    acc[0][0] = wmma_f32(a0, b0, acc[0][0]);
    acc[0][1] = wmma_f32(a0, b1, acc[0][1]);
    acc[0][2] = wmma_f32(a0, b2, acc[0][2]);
    acc[0][3] = wmma_f32(a0, b3, acc[0][3]);
    acc[1][0] = wmma_f32(a1, b0, acc[1][0]);
    acc[1][1] = wmma_f32(a1, b1, acc[1][1]);
    acc[1][2] = wmma_f32(a1, b2, acc[1][2]);
    acc[1][3] = wmma_f32(a1, b3, acc[1][3]);
  }

#pragma unroll
  for (int jr = 0; jr < 2; ++jr) {
#pragma unroll
    for (int j = 0; j < 4; ++j) {
      const int n = col0 + j * 16 + l16;
#pragma unroll
      for (int i = 0; i < 8; ++i) {
        int mr = row0 + jr * 16 + i + (half << 3);
        if (mr < N_)
          out[(size_t)bt * N_ * C_ + (size_t)mr * C_ + n] = acc[jr][j][i];
      }
    }
  }
}

// ---------------------------------------------------------------------------
// Exact jnp.percentile(70) per (b,t): stage 40000 keys in LDS, 32-step binary
// search over sortable-uint space for order stats 27999 and 28000.
// ---------------------------------------------------------------------------
__global__ void percentile_kernel(const float* __restrict__ adj, float* __restrict__ thr)
{
  extern __shared__ unsigned skey[];   // NN_ keys, 160 KB dynamic LDS
  __shared__ unsigned c0s, c1s;
  const int bt = blockIdx.x;
  const float* p = adj + (size_t)bt * NN_;
  for (int i = threadIdx.x; i < NN_; i += blockDim.x) {
    unsigned u = __float_as_uint(p[i]);
    skey[i] = (u & 0x80000000u) ? ~u : (u | 0x80000000u);
  }
  __syncthreads();
  unsigned lo0 = 0u, hi0 = 0xFFFFFFFFu;
  unsigned lo1 = 0u, hi1 = 0xFFFFFFFFu;
  const unsigned K0 = 28000u, K1 = 28001u;   // order stats 27999 & 28000
  for (int it = 0; it < 32; ++it) {
    unsigned mid0 = lo0 + ((hi0 - lo0) >> 1);
    unsigned mid1 = lo1 + ((hi1 - lo1) >> 1);
    if (threadIdx.x == 0) { c0s = 0u; c1s = 0u; }
    __syncthreads();
    unsigned c0 = 0, c1 = 0;
    for (int i = threadIdx.x; i < NN_; i += blockDim.x) {
      unsigned k = skey[i];
      c0 += (k <= mid0);
      c1 += (k <= mid1);
    }
    atomicAdd(&c0s, c0);
    atomicAdd(&c1s, c1);
    __syncthreads();
    unsigned tc0 = c0s, tc1 = c1s;
    __syncthreads();
    if (tc0 >= K0) hi0 = mid0; else lo0 = mid0 + 1;
    if (tc1 >= K1) hi1 = mid1; else lo1 = mid1 + 1;
  }
  if (threadIdx.x == 0) {
    unsigned u0 = (lo0 & 0x80000000u) ? (lo0 ^ 0x80000000u) : ~lo0;
    unsigned u1 = (lo1 & 0x80000000u) ? (lo1 ^ 0x80000000u) : ~lo1;
    float v0 = __uint_as_float(u0), v1 = __uint_as_float(u1);
    // position = 0.7*(40000-1) = 27999.3 -> linear interpolation frac 0.3
    thr[bt] = 0.7f * v0 + 0.3f * v1;
  }
}

// ---------------------------------------------------------------------------
// BatchNorm (training stats, biased variance) helpers
// ---------------------------------------------------------------------------
__global__ void zero_kernel(float* __restrict__ p, int n) {
  int i = blockIdx.x * blockDim.x + threadIdx.x;
  if (i < n) p[i] = 0.f;
}

__global__ void bn_stats(const float* __restrict__ y, float* __restrict__ stats, int R) {
  int c = threadIdx.x;                     // C_ threads
  float s = 0.f, s2 = 0.f;
  for (int r = blockIdx.x; r < R; r += gridDim.x) {
    float v = y[(size_t)r * C_ + c];
    s += v; s2 += v * v;
  }
  atomicAdd(&stats[c], s);
  atomicAdd(&stats[C_ + c], s2);
}

// act: 1=relu, 3=gelu(exact)
__global__ void bn_apply(float* __restrict__ y, const float* __restrict__ stats,
                         const float* __restrict__ g, const float* __restrict__ b,
                         int R, int act) {
  size_t tot = (size_t)R * C_;
  float invR = 1.f / (float)R;
  for (size_t i = (size_t)blockIdx.x * blockDim.x + threadIdx.x; i < tot;
       i += (size_t)gridDim.x * blockDim.x) {
    int c = (int)(i & (C_ - 1));
    float m = stats[c] * invR;
    float var = stats[C_ + c] * invR - m * m;
    float v = (y[i] - m) * rsqrtf(var + 1e-5f) * g[c] + b[c];
    if (act == 1)      v = fmaxf(v, 0.f);
    else if (act == 3) v = 0.5f * v * (1.f + erff(v * 0.7071067811865475f));
    y[i] = v;
  }
}

// ---------------------------------------------------------------------------
// SERO readout helpers
// ---------------------------------------------------------------------------
__global__ void mean_nodes(const float* __restrict__ hb, float* __restrict__ xr) {
  int tb = blockIdx.x;                 // t*B + b
  int t = tb / B_, b = tb % B_;
  int c = threadIdx.x;
  const float* base = hb + ((size_t)(b * T_ + t) * N_) * C_ + c;
  float s = 0.f;
  for (int n = 0; n < N_; ++n) s += base[(size_t)n * C_];
  xr[(size_t)tb * C_ + c] = s * (1.f / N_);
}

__global__ void wmean_nodes(const float* __restrict__ hb, const float* __restrict__ ga,
                            float* __restrict__ hr) {
  int tb = blockIdx.x;
  int t = tb / B_, b = tb % B_;
  int c = threadIdx.x;
  const float* base = hb + ((size_t)(b * T_ + t) * N_) * C_ + c;
  const float* gw = ga + (size_t)tb * N_;
  float s = 0.f;
  for (int n = 0; n < N_; ++n) s += base[(size_t)n * C_] * gw[n];
  hr[(size_t)tb * C_ + c] = s * (1.f / N_);
}

__global__ void node_out(const float* __restrict__ ga, float* __restrict__ out, int layer) {
  int i = blockIdx.x * blockDim.x + threadIdx.x;
  if (i >= RTB_ * N_) return;
  int n = i % N_;
  int tb = i / N_;
  int t = tb / B_, b = tb % B_;
  out[OUT_NODE + (((size_t)b * L_ + layer) * T_ + t) * N_ + n] = ga[i];
}

// ---------------------------------------------------------------------------
// Transformer helpers
// ---------------------------------------------------------------------------
__global__ void attn_softmax(const float* __restrict__ qkv, float* __restrict__ Abuf,
                             float* __restrict__ out, int layer) {
  int b = blockIdx.x / T_, t = blockIdx.x % T_;
  int s = threadIdx.x;                 // T_ threads
  __shared__ float sm[T_];
  __shared__ float mx, sum;
  const float* q = qkv + ((size_t)(t * B_ + b)) * (3 * C_);
  const float* k = qkv + ((size_t)(s * B_ + b)) * (3 * C_) + C_;
  float d = 0.f;
  for (int c = 0; c < C_; ++c) d += q[c] * k[c];
  sm[s] = d * 0.08838834764831845f;    // 1/sqrt(128)
  __syncthreads();
  if (s == 0) {
    float m = sm[0];
    for (int i = 1; i < T_; ++i) m = fmaxf(m, sm[i]);
    mx = m;
  }
  __syncthreads();
  float e = __expf(sm[s] - mx);
  sm[s] = e;
  __syncthreads();
  if (s == 0) {
    float ss = 0.f;
    for (int i = 0; i < T_; ++i) ss += sm[i];
    sum = ss;
  }
  __syncthreads();
  float p = e / sum;
  Abuf[((size_t)b * T_ + t) * T_ + s] = p;
  out[OUT_TIME + (((size_t)b * L_ + layer) * T_ + t) * T_ + s] = p;
}

__global__ void attn_apply(const float* __restrict__ Abuf, const float* __restrict__ qkv,
                           float* __restrict__ att) {
  int tb = blockIdx.x;
  int t = tb / B_, b = tb % B_;
  int c = threadIdx.x;
  float s = 0.f;
  for (int k = 0; k < T_; ++k)
    s += Abuf[((size_t)b * T_ + t) * T_ + k] *
         qkv[((size_t)(k * B_ + b)) * (3 * C_) + 2 * C_ + c];
  att[(size_t)tb * C_ + c] = s;
}

__global__ void layernorm(const float* __restrict__ x, const float* __restrict__ xadd,
                          const float* __restrict__ g, const float* __restrict__ b,
                          float* __restrict__ y) {
  int r = blockIdx.x;
  int c = threadIdx.x;                 // C_ threads
  __shared__ float red[C_];
  float v = x[(size_t)r * C_ + c];
  if (xadd) v += xadd[(size_t)r * C_ + c];
  red[c] = v;
  __syncthreads();
  for (int o = C_ / 2; o > 0; o >>= 1) { if (c < o) red[c] += red[c + o]; __syncthreads(); }
  float m = red[0] * (1.f / C_);
  __syncthreads();
  red[c] = (v - m) * (v - m);
  __syncthreads();
  for (int o = C_ / 2; o > 0; o >>= 1) { if (c < o) red[c] += red[c + o]; __syncthreads(); }
  float var = red[0] * (1.f / C_);
  y[(size_t)r * C_ + c] = (v - m) * rsqrtf(var + 1e-5f) * g[c] + b[c];
}

__global__ void lat_sum(const float* __restrict__ x2, float* __restrict__ lat) {
  int b = blockIdx.x;                  // B_ blocks
  int c = threadIdx.x;
  float s = 0.f;
  for (int t = 0; t < T_; ++t) s += x2[((size_t)(t * B_ + b)) * C_ + c];
  lat[b * C_ + c] = s;
}

__global__ void cls_accum(const float* __restrict__ lat, const float* __restrict__ cw,
                          const float* __restrict__ cb, float* __restrict__ out) {
  int i = threadIdx.x;
  if (i >= B_ * NC_) return;
  int b = i / NC_, k = i % NC_;
  float s = cb[k];
  for (int c = 0; c < C_; ++c) s += lat[b * C_ + c] * cw[c * NC_ + k];
  out[OUT_LOGIT + b * NC_ + k] += s;
}

__global__ void feat_accum(const float* __restrict__ lat, float* __restrict__ out) {
  int i = blockIdx.x * blockDim.x + threadIdx.x;
  if (i < B_ * C_) out[OUT_FFMRI + i] += lat[i] * (1.f / L_);
}

__global__ void smri_kernel(const float* __restrict__ trad, const float* __restrict__ w,
                            const float* __restrict__ bias, const float* __restrict__ g,
                            const float* __restrict__ bb, float* __restrict__ out) {
  int c = threadIdx.x;                 // C_ threads
  float y[B_];
  float m = 0.f;
  for (int b = 0; b < B_; ++b) {
    float s = bias[c];
    for (int k = 0; k < S_; ++k) s += trad[b * S_ + k] * w[k * C_ + c];
    y[b] = s; m += s;
  }
  m *= (1.f / B_);
  float var = 0.f;
  for (int b = 0; b < B_; ++b) var += (y[b] - m) * (y[b] - m);
  var *= (1.f / B_);
  float inv = rsqrtf(var + 1e-5f);
  for (int b = 0; b < B_; ++b)
    out[OUT_FSMRI + b * C_ + c] = (y[b] - m) * inv * g[c] + bb[c];
}

// ---------------------------------------------------------------------------
// Host driver
// ---------------------------------------------------------------------------
extern "C" void kernel_launch(void* const* d_in, const int* in_sizes, int n_in,
                              void* d_out, int out_size, void* d_ws, size_t ws_size,
                              hipStream_t stream) {
  (void)in_sizes; (void)n_in; (void)out_size; (void)ws_size;
  const float* fMRI_v    = (const float*)d_in[0];
  const float* fMRI_a    = (const float*)d_in[1];
  const float* sMRI_trad = (const float*)d_in[2];
  /* d_in[3] sMRI_deep, d_in[4] csatt, d_in[5] CA_MOD: unused by reference */
  const float* w_init    = (const float*)d_in[6];
  const float* b_init    = (const float*)d_in[7];
  const float* gin_eps   = (const float*)d_in[8];
  const float* gin_w1    = (const float*)d_in[9];
  const float* gin_b1    = (const float*)d_in[10];
  const float* gin_bn1_g = (const float*)d_in[11];
  const float* gin_bn1_b = (const float*)d_in[12];
  const float* gin_w2    = (const float*)d_in[13];
  const float* gin_b2    = (const float*)d_in[14];
  const float* gin_bn2_g = (const float*)d_in[15];
  const float* gin_bn2_b = (const float*)d_in[16];
  const float* sero_w    = (const float*)d_in[17];
  const float* sero_b    = (const float*)d_in[18];
  const float* sero_bn_g = (const float*)d_in[19];
  const float* sero_bn_b = (const float*)d_in[20];
  const float* sero_aw   = (const float*)d_in[21];
  const float* sero_ab   = (const float*)d_in[22];
  const float* tr_wqkv   = (const float*)d_in[23];
  const float* tr_bqkv   = (const float*)d_in[24];
  const float* tr_wo     = (const float*)d_in[25];
  const float* tr_bo     = (const float*)d_in[26];
  const float* tr_ln1_g  = (const float*)d_in[27];
  const float* tr_ln1_b  = (const float*)d_in[28];
  const float* tr_ln2_g  = (const float*)d_in[29];
  const float* tr_ln2_b  = (const float*)d_in[30];
  const float* tr_w1     = (const float*)d_in[31];
  const float* tr_b1     = (const float*)d_in[32];
  const float* tr_w2     = (const float*)d_in[33];
  const float* tr_b2     = (const float*)d_in[34];
  const float* cls_w     = (const float*)d_in[35];
  const float* cls_b     = (const float*)d_in[36];
  const float* smri_w    = (const float*)d_in[37];
  const float* smri_b    = (const float*)d_in[38];
  const float* smri_bn_g = (const float*)d_in[39];
  const float* smri_bn_b = (const float*)d_in[40];

  float* out = (float*)d_out;
  float* ws  = (float*)d_ws;
  size_t cur = 0;
  auto alloc = [&](size_t n) { float* p = ws + cur; cur += n; return p; };
  float* h3     = alloc((size_t)RTOK_ * C_);
  float* aggbuf = alloc((size_t)RTOK_ * C_);
  float* buf1   = alloc((size_t)RTOK_ * C_);
  float* thr    = alloc(B_ * T_);
  float* stats  = alloc(2 * C_);
  float* xr     = alloc((size_t)RTB_ * C_);
  float* ebuf   = alloc((size_t)RTB_ * C_);
  float* gabuf  = alloc((size_t)RTB_ * N_);
  float* hr     = alloc((size_t)RTB_ * C_);
  float* qkv    = alloc((size_t)RTB_ * 3 * C_);
  float* Abuf   = alloc((size_t)B_ * T_ * T_);
  float* att    = alloc((size_t)RTB_ * C_);
  float* attp   = alloc((size_t)RTB_ * C_);
  float* x1t    = alloc((size_t)RTB_ * C_);
  float* mbuf   = alloc((size_t)RTB_ * 2 * C_);
  float* m2     = alloc((size_t)RTB_ * C_);
  float* x2t    = alloc((size_t)RTB_ * C_);
  float* lat    = alloc((size_t)B_ * C_);

  // zero accumulated output regions (logit + feat_fMRI)
  zero_kernel<<<5, 256, 0, stream>>>(out, OUT_FSMRI);

  // h3 = fMRI_v @ w_init + b_init   (102400 x 200 x 128)
  dim3 g0(C_ / 64, RTOK_ / 32);
  gemm_wmma<CIN_, C_, 0><<<g0, 32, 0, stream>>>(fMRI_v, w_init, b_init, h3, RTOK_);

  // per-(b,t) percentile threshold
  percentile_kernel<<<B_ * T_, 256, NN_ * sizeof(unsigned), stream>>>(fMRI_a, thr);

  dim3 g1(C_ / 64, RTOK_ / 32);            // 102400 x 128 x 128
  dim3 gs(C_ / 64, RTB_ / 32);             // 512 x 128 x 128
  for (int l = 0; l < L_; ++l) {
    // agg = (mask + eps*I) @ h3
    dim3 gag(C_ / 64, (N_ + 31) / 32, B_ * T_);
    agg_wmma<<<gag, 32, 0, stream>>>(fMRI_a, thr, h3, gin_eps, l, aggbuf);

    // GIN MLP1 + BN + relu
    gemm_wmma<C_, C_, 0><<<g1, 32, 0, stream>>>(aggbuf, gin_w1 + (size_t)l * C_ * C_,
                                                gin_b1 + l * C_, buf1, RTOK_);
    zero_kernel<<<1, 256, 0, stream>>>(stats, 2 * C_);
    bn_stats<<<240, C_, 0, stream>>>(buf1, stats, RTOK_);
    bn_apply<<<1024, 256, 0, stream>>>(buf1, stats, gin_bn1_g + l * C_,
                                       gin_bn1_b + l * C_, RTOK_, 1);
    // GIN MLP2 + BN + relu -> hb (in aggbuf)
    gemm_wmma<C_, C_, 0><<<g1, 32, 0, stream>>>(buf1, gin_w2 + (size_t)l * C_ * C_,
                                                gin_b2 + l * C_, aggbuf, RTOK_);
    zero_kernel<<<1, 256, 0, stream>>>(stats, 2 * C_);
    bn_stats<<<240, C_, 0, stream>>>(aggbuf, stats, RTOK_);
    bn_apply<<<1024, 256, 0, stream>>>(aggbuf, stats, gin_bn2_g + l * C_,
                                       gin_bn2_b + l * C_, RTOK_, 1);

    // SERO readout
    mean_nodes<<<RTB_, C_, 0, stream>>>(aggbuf, xr);
    gemm_wmma<C_, C_, 0><<<gs, 32, 0, stream>>>(xr, sero_w + (size_t)l * C_ * C_,
                                                sero_b + l * C_, ebuf, RTB_);
    zero_kernel<<<1, 256, 0, stream>>>(stats, 2 * C_);
    bn_stats<<<64, C_, 0, stream>>>(ebuf, stats, RTB_);
    bn_apply<<<64, 256, 0, stream>>>(ebuf, stats, sero_bn_g + l * C_,
                                     sero_bn_b + l * C_, RTB_, 3);   // gelu
    dim3 gsa((N_ + 63) / 64, RTB_ / 32);   // 512 x 128 x 200, sigmoid
    gemm_wmma<C_, N_, 2><<<gsa, 32, 0, stream>>>(ebuf, sero_aw + (size_t)l * C_ * N_,
                                                 sero_ab + l * N_, gabuf, RTB_);
    node_out<<<(RTB_ * N_ + 255) / 256, 256, 0, stream>>>(gabuf, out, l);
    wmean_nodes<<<RTB_, C_, 0, stream>>>(aggbuf, gabuf, hr);

    // Transformer
    dim3 gq((3 * C_) / 64, RTB_ / 32);     // 512 x 128 x 384
    gemm_wmma<C_, 3 * C_, 0><<<gq, 32, 0, stream>>>(hr, tr_wqkv + (size_t)l * C_ * 3 * C_,
                                                    tr_bqkv + l * 3 * C_, qkv, RTB_);
    attn_softmax<<<B_ * T_, T_, 0, stream>>>(qkv, Abuf, out, l);
    attn_apply<<<RTB_, C_, 0, stream>>>(Abuf, qkv, att);
    gemm_wmma<C_, C_, 0><<<gs, 32, 0, stream>>>(att, tr_wo + (size_t)l * C_ * C_,
                                                tr_bo + l * C_, attp, RTB_);
    layernorm<<<RTB_, C_, 0, stream>>>(attp, nullptr, tr_ln1_g + l * C_,
                                       tr_ln1_b + l * C_, x1t);
    dim3 gf1((2 * C_) / 64, RTB_ / 32);    // 512 x 128 x 256, relu
    gemm_wmma<C_, 2 * C_, 1><<<gf1, 32, 0, stream>>>(x1t, tr_w1 + (size_t)l * C_ * 2 * C_,
                                                     tr_b1 + l * 2 * C_, mbuf, RTB_);
    gemm_wmma<2 * C_, C_, 0><<<gs, 32, 0, stream>>>(mbuf, tr_w2 + (size_t)l * 2 * C_ * C_,
                                                    tr_b2 + l * C_, m2, RTB_);
    layernorm<<<RTB_, C_, 0, stream>>>(x1t, m2, tr_ln2_g + l * C_,
                                       tr_ln2_b + l * C_, x2t);

    // heads
    lat_sum<<<B_, C_, 0, stream>>>(x2t, lat);
    cls_accum<<<1, 32, 0, stream>>>(lat, cls_w + (size_t)l * C_ * NC_,
                                    cls_b + l * NC_, out);
    feat_accum<<<(B_ * C_ + 255) / 256, 256, 0, stream>>>(lat, out);
  }

  // sMRI branch
  smri_kernel<<<1, C_, 0, stream>>>(sMRI_trad, smri_w, smri_b, smri_bn_g,
                                    smri_bn_b, out);
}